// ResonanceEngine_2276332667136
// MI455X (gfx1250) — compile-verified
//
#include <hip/hip_runtime.h>

typedef __attribute__((ext_vector_type(2))) float v2f;
typedef __attribute__((ext_vector_type(8))) float v8f;

#define NN   2048   // num_nodes
#define DD   64     // dim
#define TAU  1.0f
#define BLK  256    // threads per block (8 waves)

__launch_bounds__(BLK, 1)
__global__ void resonance_softmax_kernel(const int*   __restrict__ node_idx,
                                         const float* __restrict__ ctx,
                                         const float* __restrict__ W,
                                         float*       __restrict__ out)
{
    __shared__ float energy[NN];   // 8 KB
    __shared__ float red[BLK];     // 1 KB

    const int tid  = threadIdx.x;
    const int lane = tid & 31;
    const int wave = tid >> 5;

    // Gathered slab for this query: W[idx] is (NN, DD) f32, contiguous.
    const long long sel = (long long)node_idx[blockIdx.x];
    const float* __restrict__ slab = W + sel * (long long)(NN * DD);

    // ---- B fragments: B[k][j] = ctx[d0 + k], identical for all 16 columns j.
    // B (4x16 f32) layout: lanes 0-15 hold K=0 (v0) / K=1 (v1);
    //                      lanes 16-31 hold K=2 (v0) / K=3 (v1).
    const int kb = (lane >> 4) << 1;   // 0 or 2
    v2f bfrag[16];
    #pragma unroll
    for (int kk = 0; kk < 16; ++kk) {
        bfrag[kk].x = ctx[kk * 4 + kb + 0];
        bfrag[kk].y = ctx[kk * 4 + kb + 1];
    }

    // ---- Phase 1: energies via chained f32 WMMA (16x16x4), one wave per 256 rows.
    // A (16x4 f32) layout: lane supplies row M = lane&15; lanes<16 give K=0/1,
    // lanes>=16 give K=2/3 -> per-lane float2 load at (row, d0 + kb).
    const int m = lane & 15;
    // Each lane deposits the energy for row (lane>>4)*8 + (lane&7); lanes 8-15
    // and 24-31 duplicate lanes 0-7 / 16-23 (identical data, benign LDS race).
    const int eoff = ((lane >> 4) << 3) + (lane & 7);

    #pragma unroll 1
    for (int t = 0; t < 16; ++t) {
        const int n0 = (wave * 16 + t) * 16;
        const float* __restrict__ rowp = slab + (long long)(n0 + m) * DD + kb;

        if (t < 15) __builtin_prefetch(rowp + 16 * DD, 0, 3);  // next tile

        v8f acc = {0.f, 0.f, 0.f, 0.f, 0.f, 0.f, 0.f, 0.f};
        #pragma unroll
        for (int kk = 0; kk < 16; ++kk) {
            v2f a;
            a.x = rowp[kk * 4 + 0];
            a.y = rowp[kk * 4 + 1];
            acc = __builtin_amdgcn_wmma_f32_16x16x4_f32(
                      /*neg_a=*/false, a, /*neg_b=*/false, bfrag[kk],
                      /*c_mod=*/(short)0, acc, /*reuse_a=*/false, /*reuse_b=*/false);
        }

        // All 16 columns of acc are identical. C/D layout: vgpr v holds row M=v
        // (lanes 0-15) or M=v+8 (lanes 16-31). Branch-free extract: select
        // acc[lane&7] via cndmask chain, then every lane stores (duplicates
        // write the same value to the same address).
        float val = acc[0];
        #pragma unroll
        for (int v = 1; v < 8; ++v)
            val = ((lane & 7) == v) ? acc[v] : val;
        energy[n0 + eoff] = val;
    }
    __syncthreads();

    // ---- Phase 2: softmax over the NN energies (TAU-scaled), all in LDS.
    const float inv_tau = 1.0f / TAU;

    // Row max
    float lmax = -3.402823466e+38f;
    #pragma unroll
    for (int i = 0; i < NN / BLK; ++i)
        lmax = fmaxf(lmax, energy[tid + i * BLK]);
    red[tid] = lmax;
    __syncthreads();
    for (int s = BLK / 2; s > 0; s >>= 1) {
        if (tid < s) red[tid] = fmaxf(red[tid], red[tid + s]);
        __syncthreads();
    }
    const float gmax = red[0];
    __syncthreads();

    // Exponentials + sum
    float ev[NN / BLK];
    float lsum = 0.f;
    #pragma unroll
    for (int i = 0; i < NN / BLK; ++i) {
        ev[i] = __expf((energy[tid + i * BLK] - gmax) * inv_tau);
        lsum += ev[i];
    }
    red[tid] = lsum;
    __syncthreads();
    for (int s = BLK / 2; s > 0; s >>= 1) {
        if (tid < s) red[tid] += red[tid + s];
        __syncthreads();
    }
    const float inv_sum = 1.0f / red[0];

    float* __restrict__ outb = out + (long long)blockIdx.x * NN;
    #pragma unroll
    for (int i = 0; i < NN / BLK; ++i)
        outb[tid + i * BLK] = ev[i] * inv_sum;
}

extern "C" void kernel_launch(void* const* d_in, const int* in_sizes, int n_in,
                              void* d_out, int out_size, void* d_ws, size_t ws_size,
                              hipStream_t stream) {
    (void)n_in; (void)out_size; (void)d_ws; (void)ws_size;
    const int*   node_idx = (const int*)d_in[0];    // (B,) indices
    const float* ctx      = (const float*)d_in[1];  // (D,) f32
    const float* W        = (const float*)d_in[2];  // (N, N, D) f32
    float*       out      = (float*)d_out;          // (B, N) f32

    const int Bq = in_sizes[0];                     // 256 queries -> 256 blocks
    resonance_softmax_kernel<<<Bq, BLK, 0, stream>>>(node_idx, ctx, W, out);
}